// v_LPTC_13795434955466
// MI455X (gfx1250) — compile-verified
//
#include <hip/hip_runtime.h>

typedef __attribute__((ext_vector_type(2))) float v2f;
typedef __attribute__((ext_vector_type(4))) float v4f;
typedef __attribute__((ext_vector_type(8))) float v8f;

#define NSITE 32768

// Layouts (complex64 stored as interleaved float pairs):
//   feat : [i(2)][site][t(4)][g(3)]
//   wts  : [i(2)][o(2)][p(9)][site][s(4)][t(4)]
//   U    : [mu(4)][site][a(3)][b(3)]
//   out  : [o(2)][site][s(4)][g(3)]
__global__ __launch_bounds__(128) void lptc_wmma_kernel(
    const float* __restrict__ feat,
    const float* __restrict__ wts,
    const float* __restrict__ U,
    float* __restrict__ out)
{
    // Per-wave B-fragment staging: 18 K-groups x 32 lanes x 2 VGPR slots.
    // Slot [kg][L][v] holds B[row=(L>>4)*2+v][col=L&15] of the 4x16 fragment.
    __shared__ float Bstage[4][18 * 64];
    __shared__ float Cbuf[4][16 * 6];

    const int lane = threadIdx.x & 31;
    const int wave = threadIdx.x >> 5;
    const int site = blockIdx.x * 4 + wave;
    const int m    = lane & 15;

    // site bits: z[3:0] y[7:4] x[11:8] t[14:12]
    const int sz = site & 15;
    const int sy = (site >> 4) & 15;
    const int sx = (site >> 8) & 15;
    const int st = site >> 12;

    int nf[4], nb[4]; // periodic neighbors per lattice axis
    nf[0] = (((st + 1) & 7) << 12) | (site & 0x0FFF);
    nb[0] = (((st + 7) & 7) << 12) | (site & 0x0FFF);
    nf[1] = (site & ~0x0F00) | (((sx + 1) & 15) << 8);
    nb[1] = (site & ~0x0F00) | (((sx + 15) & 15) << 8);
    nf[2] = (site & ~0x00F0) | (((sy + 1) & 15) << 4);
    nb[2] = (site & ~0x00F0) | (((sy + 15) & 15) << 4);
    nf[3] = (site & ~0x000F) | ((sz + 1) & 15);
    nb[3] = (site & ~0x000F) | ((sz + 15) & 15);

    float* Bst = &Bstage[wave][0];

    // ---------- Phase 1: transported features written straight into B-fragment slots ----------
    // Invalid fragment columns (6..15 of each lane half) are zeroed by their owner lanes;
    // valid columns (0..5 / 16..21) are written by the 24 compute lanes. Disjoint -> one barrier.
    if (m >= 6) {
        const v2f z = {0.f, 0.f};
        #pragma unroll
        for (int kg = 0; kg < 18; ++kg)
            *(v2f*)&Bst[kg * 64 + lane * 2] = z;
    }
    if (lane < 24) {
        const int li = lane / 12;          // input feature
        const int lt = (lane % 12) / 3;    // spin (K row within group)
        const int lg = lane % 3;           // output color (fragment column)

        // float index within a K-group region: col*2 + vgpr
        const int cRe = ((lt >> 1) * 16 + lg) * 2 + (lt & 1);
        const int cIm = cRe + 6;           // +3 columns

        const v2f* fv = (const v2f*)feat;
        const v2f* uv = (const v2f*)U;

        // p = 0 : identity  (kg = li)
        {
            v2f v = fv[((li * NSITE + site) * 4 + lt) * 3 + lg];
            Bst[li * 64 + cRe] = v.x;
            Bst[li * 64 + cIm] = v.y;
        }
        #pragma unroll
        for (int mu = 0; mu < 4; ++mu) {
            // fwd (p = 1+2mu): sum_b U_mu(x)[g][b] * v(x+mu)[t][b]
            float ar = 0.f, ai = 0.f;
            #pragma unroll
            for (int b = 0; b < 3; ++b) {
                v2f u = uv[((mu * NSITE + site) * 3 + lg) * 3 + b];
                v2f v = fv[((li * NSITE + nf[mu]) * 4 + lt) * 3 + b];
                ar += u.x * v.x - u.y * v.y;
                ai += u.x * v.y + u.y * v.x;
            }
            int kg = (1 + 2 * mu) * 2 + li;
            Bst[kg * 64 + cRe] = ar;
            Bst[kg * 64 + cIm] = ai;

            // bwd (p = 2+2mu): sum_b conj(U_mu(x-mu)[b][g]) * v(x-mu)[t][b]
            ar = 0.f; ai = 0.f;
            #pragma unroll
            for (int b = 0; b < 3; ++b) {
                v2f u = uv[((mu * NSITE + nb[mu]) * 3 + b) * 3 + lg];
                v2f v = fv[((li * NSITE + nb[mu]) * 4 + lt) * 3 + b];
                ar += u.x * v.x + u.y * v.y;
                ai += u.x * v.y - u.y * v.x;
            }
            kg = (2 + 2 * mu) * 2 + li;
            Bst[kg * 64 + cRe] = ar;
            Bst[kg * 64 + cIm] = ai;
        }
    }
    __syncthreads();

    // ---------- Phase 2: WMMA chain, K = 72 over (p,i,t), 18 x 16x16x4 f32 ----------
    // A rows 0-7: Re W[(o,s)], rows 8-15: Im W ; B cols 0-2: Re T[g], 3-5: Im T[g]
    const int khalf = lane >> 4;        // 0 -> K{0,1}, 1 -> K{2,3}
    const int tbase = khalf * 2;
    const int os    = m & 7;
    const int o_    = os >> 2;
    const int s_    = os & 3;
    const bool isIm = m >= 8;

    v8f c = {0.f, 0.f, 0.f, 0.f, 0.f, 0.f, 0.f, 0.f};
    const v4f* wp = (const v4f*)wts;

    #pragma unroll
    for (int p = 0; p < 9; ++p) {
        #pragma unroll
        for (int i = 0; i < 2; ++i) {
            // A fragment: one contiguous b128 (two consecutive complex t) per lane,
            // streamed nontemporally (weights are single-use, dominant 151 MB traffic)
            int widx = ((((i * 2 + o_) * 9 + p) * NSITE + site) * 4 + s_) * 4 + tbase;
            v4f w = __builtin_nontemporal_load(wp + (widx >> 1));
            v2f a;
            a.x = isIm ? w.y : w.x;   // K = tbase
            a.y = isIm ? w.w : w.z;   // K = tbase + 1

            // B fragment: single unconditional b64, pre-staged in fragment layout
            v2f b = *(const v2f*)&Bst[(p * 2 + i) * 64 + lane * 2];

            c = __builtin_amdgcn_wmma_f32_16x16x4_f32(
                    false, a, false, b, (short)0, c, false, false);
        }
    }

    // ---------- Phase 3: recombine real blocks into complex output ----------
    // out_re = C[0:8,0:3] - C[8:16,3:6] ; out_im = C[0:8,3:6] + C[8:16,0:3]
    float* Cb = &Cbuf[wave][0];
    if (m < 6) {
        const int rbase = khalf * 8;   // lanes 16-31 hold rows 8-15
        #pragma unroll
        for (int r = 0; r < 8; ++r)
            Cb[(rbase + r) * 6 + m] = c[r];
    }
    __syncthreads();

    if (lane < 24) {
        const int os2 = lane / 3;
        const int g2  = lane % 3;
        float re = Cb[os2 * 6 + g2]     - Cb[(os2 + 8) * 6 + g2 + 3];
        float im = Cb[os2 * 6 + g2 + 3] + Cb[(os2 + 8) * 6 + g2];
        const int o2 = os2 >> 2, s2 = os2 & 3;
        const int oidx = ((o2 * NSITE + site) * 4 + s2) * 3 + g2;
        out[2 * oidx]     = re;
        out[2 * oidx + 1] = im;
    }
}

extern "C" void kernel_launch(void* const* d_in, const int* in_sizes, int n_in,
                              void* d_out, int out_size, void* d_ws, size_t ws_size,
                              hipStream_t stream) {
    const float* feat = (const float*)d_in[0];  // features_in (complex64)
    const float* wts  = (const float*)d_in[1];  // weights     (complex64)
    const float* U    = (const float*)d_in[2];  // gauge links (complex64)
    float* out = (float*)d_out;                 // [2][NSITE][4][3] complex64
    lptc_wmma_kernel<<<NSITE / 4, 128, 0, stream>>>(feat, wts, U, out);
}